// LMSA_72877005078542
// MI455X (gfx1250) — compile-verified
//
#include <hip/hip_runtime.h>

// LMSA collapses: softmax over broadcast-constant rows == uniform 1/N, so
// attn = 1/1024 exactly; convs/sigmoid are dead code. Remaining work:
//   V = X @ Wv^T           (fp32 WMMA 16x16x4)
//   mean[b,k] = mean of flat-V block k (1024 contiguous elems; torch-view trick)
//   y[b,:] = Wp @ mean[b,:] + bp ; out[b,n,:] = y[b,:] for all n
// lcm(1024,192)=3072 = one 16-row M-tile => each M-tile owns exactly 3 blocks.

typedef __attribute__((ext_vector_type(2))) float v2f;
typedef __attribute__((ext_vector_type(8))) float v8f;

#define DIMC   192
#define NTOK   1024        // L*L
#define BATCH  8
#define NPB    196608      // NTOK*DIMC flat floats per batch

// ---------------------------------------------------------------------------
// Kernel 1: V-GEMM (WMMA f32 16x16x4) fused with per-block mean reduction.
// grid = BATCH*64 (one block per (b, 16-row M-tile)), blockDim = 384 (12 waves,
// one wave per 16-col N-tile covering all 192 output channels).
// ---------------------------------------------------------------------------
__global__ __launch_bounds__(384)
void lmsa_vmean_wmma(const float* __restrict__ x,
                     const float* __restrict__ Wv,
                     float* __restrict__ mean /* [BATCH*192] */) {
    const int lane = threadIdx.x & 31;
    const int wave = threadIdx.x >> 5;        // 0..11 -> N-tile
    const int b    = blockIdx.x >> 6;
    const int mt   = blockIdx.x & 63;         // M-tile within batch
    const int r0   = mt << 4;                 // first token row of tile
    const int n0   = wave << 4;               // first output channel of tile
    const int lm   = lane & 15;
    const int kh   = (lane >> 4) << 1;        // lanes 0-15: K{0,1}; 16-31: K{2,3}

    // A[m,k] = x[b, r0+m, k]   (row-major [B,N,C])
    const float* Ap = x  + (size_t)(b * NTOK + r0 + lm) * DIMC + kh;
    // B[k,n] = Wv[n0+n, k]     (v[.,d] = sum_c x[.,c]*Wv[d,c])
    const float* Bp = Wv + (size_t)(n0 + lm) * DIMC + kh;

    v8f acc = {};
#pragma unroll 8
    for (int k0 = 0; k0 < DIMC; k0 += 4) {
        v2f af = *(const v2f*)(Ap + k0);    // 8B-aligned b64 load
        v2f bf = *(const v2f*)(Bp + k0);
        acc = __builtin_amdgcn_wmma_f32_16x16x4_f32(
            /*neg_a=*/false, af, /*neg_b=*/false, bf,
            /*c_mod=*/(short)0, acc, /*reuse_a=*/false, /*reuse_b=*/false);
    }

    // C layout: VGPR j holds element (row = j + 8*(lane>=16), col = lane&15).
    // Flat-V index inside this M-tile: f = row*192 + (n0+col), f in [0,3072),
    // spanning exactly blocks 3*mt .. 3*mt+2 (boundaries at f=1024, 2048).
    // Branchless partition: s2 = total - s0 - s1 (exact: disjoint selects).
    const int col = n0 + lm;
    const int hi  = (lane >> 4) << 3;
    float s0 = 0.f, s1 = 0.f, st = 0.f;
#pragma unroll
    for (int j = 0; j < 8; ++j) {
        const int f   = (j + hi) * DIMC + col;
        const float v = acc[j];
        st += v;
        s0 += (f < 1024) ? v : 0.0f;                  // v_cmp + v_cndmask
        s1 += (f >= 1024 && f < 2048) ? v : 0.0f;
    }
    float s2 = st - s0 - s1;

    // Wave-level butterfly reduction (lane-permute HW, no LDS traffic) ...
#pragma unroll
    for (int off = 16; off >= 1; off >>= 1) {
        s0 += __shfl_xor(s0, off, 32);
        s1 += __shfl_xor(s1, off, 32);
        s2 += __shfl_xor(s2, off, 32);
    }

    // ... then one ds_add_f32 per (wave, block): 36 atomics vs 1152 before.
    __shared__ float bsum[3];
    if (threadIdx.x < 3) bsum[threadIdx.x] = 0.f;
    __syncthreads();
    if (lane == 0) {
        atomicAdd(&bsum[0], s0);
        atomicAdd(&bsum[1], s1);
        atomicAdd(&bsum[2], s2);
    }
    __syncthreads();
    if (threadIdx.x < 3)
        mean[(size_t)b * DIMC + 3 * mt + threadIdx.x] =
            bsum[threadIdx.x] * (1.0f / 1024.0f);
}

// ---------------------------------------------------------------------------
// Kernel 2: y[b,d] = sum_k Wp[d,k] * mean[b,k] + bp[d].   grid=8, block=192.
// ---------------------------------------------------------------------------
__global__ __launch_bounds__(192)
void lmsa_proj(const float* __restrict__ mean,
               const float* __restrict__ Wp,
               const float* __restrict__ bp,
               float* __restrict__ y /* [BATCH*192] */) {
    const int d = threadIdx.x;
    const int b = blockIdx.x;
    const float* m  = mean + (size_t)b * DIMC;
    const float* wr = Wp + (size_t)d * DIMC;
    float acc = bp[d];
#pragma unroll 4
    for (int k = 0; k < DIMC; ++k) acc = fmaf(wr[k], m[k], acc);
    y[(size_t)b * DIMC + d] = acc;
}

// ---------------------------------------------------------------------------
// Kernel 3: broadcast y over all 1024 token rows with 128-bit stores.
// grid = 1536, block = 256 -> 393216 float4 = 1,572,864 floats.
// ---------------------------------------------------------------------------
__global__ __launch_bounds__(256)
void lmsa_bcast(const float* __restrict__ y, float* __restrict__ out) {
    int gid = blockIdx.x * 256 + threadIdx.x;
    int f   = gid << 2;                       // flat output index (multiple of 4)
    int b   = f / NPB;
    int rem = f - b * NPB;
    int d   = rem % DIMC;                     // 192 % 4 == 0 -> stays in-row
    float4 v = *(const float4*)(y + b * DIMC + d);
    *(float4*)(out + f) = v;
}

// ---------------------------------------------------------------------------
extern "C" void kernel_launch(void* const* d_in, const int* in_sizes, int n_in,
                              void* d_out, int out_size, void* d_ws, size_t ws_size,
                              hipStream_t stream) {
    (void)in_sizes; (void)n_in; (void)out_size; (void)ws_size;
    const float* x  = (const float*)d_in[0];
    // d_in[1..5] (att_token, Wq, bq, Wk, bk) are provably dead: softmax of a
    // row-constant logit matrix is uniform regardless of the gate values.
    const float* Wv = (const float*)d_in[6];
    const float* Wp = (const float*)d_in[7];
    const float* bp = (const float*)d_in[8];
    float* out = (float*)d_out;

    float* wsf  = (float*)d_ws;
    float* mean = wsf;                 // BATCH*192 floats
    float* yvec = wsf + BATCH * DIMC;  // BATCH*192 floats

    lmsa_vmean_wmma<<<dim3(BATCH * 64), dim3(384), 0, stream>>>(x, Wv, mean);
    lmsa_proj     <<<dim3(BATCH),      dim3(192), 0, stream>>>(mean, Wp, bp, yvec);
    lmsa_bcast    <<<dim3(1536),       dim3(256), 0, stream>>>(yvec, out);
}